// HeteroGNN_87677462380649
// MI455X (gfx1250) — compile-verified
//
#include <hip/hip_runtime.h>

#define N_EL 200000
#define N_GRPN 50000
#define NE 800000
#define DIM 128

typedef float v2f __attribute__((ext_vector_type(2)));
typedef float v8f __attribute__((ext_vector_type(8)));

// ---------------- scatter (segment-sum) kernels ----------------

__global__ void scatter_cnt_kernel(const int* __restrict__ edge,
                                   float* __restrict__ cnt) {
    int e = blockIdx.x * blockDim.x + threadIdx.x;
    if (e < NE) atomicAdd(&cnt[edge[NE + e]], 1.0f);
}

// 32 lanes per edge; each lane moves 4 consecutive floats (128 = 32*4)
__global__ void scatter_feat_kernel(const int* __restrict__ edge,
                                    const float* __restrict__ xsrc,
                                    float* __restrict__ msum) {
    int idx = blockIdx.x * blockDim.x + threadIdx.x;
    int e = idx >> 5;
    int lane = idx & 31;
    if (e >= NE) return;
    int s = edge[e];
    int d = edge[NE + e];
    const float4 v = ((const float4*)(xsrc + (size_t)s * DIM))[lane];
    float* dp = msum + (size_t)d * DIM + lane * 4;
    atomicAdd(dp + 0, v.x);
    atomicAdd(dp + 1, v.y);
    atomicAdd(dp + 2, v.z);
    atomicAdd(dp + 3, v.w);
}

// ---------------- fused WMMA GEMM ----------------
// C[row, :] (+)= (A1[row,:] * invcnt) @ W1 + A2[row,:] @ W2 + bias   (FUSED)
// C[row, :]  =   A1[row,:] @ W1 + bias                               (!FUSED)
// 32 rows per block (two 16-row tiles), 4 wave32 per block; each wave owns 32
// output columns (two 16x16 N-tiles) for both row tiles -> B fragments feed
// 4 WMMAs per K-step. HAS2/FUSED/ACCUM are compile-time so the K-loop is
// straight-line (no per-iteration scalar branches); EXEC is all-ones at WMMAs.

template <bool FUSED, bool ACCUM, bool HAS2>
__device__ __forceinline__ void gemm_tile_body(
    const float* __restrict__ A1, const float* __restrict__ cnt,
    const float* __restrict__ A2, const float* __restrict__ W1,
    const float* __restrict__ W2, const float* __restrict__ bias,
    float* __restrict__ C, int base0) {
    const int lane = threadIdx.x & 31;
    const int wv   = threadIdx.x >> 5;
    const int half = lane >> 4;    // K-pair selector for A/B fragments
    const int l16  = lane & 15;    // M for A-frag, N for B/C-frags
    const int base1 = base0 + 16;
    const int row0 = base0 + l16;
    const int row1 = base1 + l16;

    float inv0 = 1.0f, inv1 = 1.0f;
    if constexpr (FUSED) {
        inv0 = 1.0f / fmaxf(cnt[row0], 1.0f);
        if constexpr (HAS2) inv1 = 1.0f / fmaxf(cnt[row1], 1.0f);
    }

    v8f acc00 = {0.f,0.f,0.f,0.f,0.f,0.f,0.f,0.f};
    v8f acc01 = {0.f,0.f,0.f,0.f,0.f,0.f,0.f,0.f};
    v8f acc10 = {0.f,0.f,0.f,0.f,0.f,0.f,0.f,0.f};
    v8f acc11 = {0.f,0.f,0.f,0.f,0.f,0.f,0.f,0.f};

    const int n0 = wv * 32 + l16;
    const float* arow0 = A1 + (size_t)row0 * DIM;
    const float* arow1 = A1 + (size_t)row1 * DIM;

#pragma unroll 4
    for (int k0 = 0; k0 < DIM; k0 += 4) {
        const int kk = k0 + half * 2;
        v2f b0, b1;
        b0.x = W1[kk * DIM + n0];
        b0.y = W1[(kk + 1) * DIM + n0];
        b1.x = W1[kk * DIM + n0 + 16];
        b1.y = W1[(kk + 1) * DIM + n0 + 16];
        v2f a0;
        a0.x = arow0[kk];
        a0.y = arow0[kk + 1];
        if constexpr (FUSED) { a0.x *= inv0; a0.y *= inv0; }
        acc00 = __builtin_amdgcn_wmma_f32_16x16x4_f32(false, a0, false, b0,
                                                      (short)0, acc00, false, false);
        acc01 = __builtin_amdgcn_wmma_f32_16x16x4_f32(false, a0, false, b1,
                                                      (short)0, acc01, false, false);
        if constexpr (HAS2) {
            v2f a1;
            a1.x = arow1[kk];
            a1.y = arow1[kk + 1];
            if constexpr (FUSED) { a1.x *= inv1; a1.y *= inv1; }
            acc10 = __builtin_amdgcn_wmma_f32_16x16x4_f32(false, a1, false, b0,
                                                          (short)0, acc10, false, false);
            acc11 = __builtin_amdgcn_wmma_f32_16x16x4_f32(false, a1, false, b1,
                                                          (short)0, acc11, false, false);
        }
    }

    if constexpr (FUSED) {  // second K=128 half: x_dst @ W2 (no scaling)
        const float* brow0 = A2 + (size_t)row0 * DIM;
        const float* brow1 = A2 + (size_t)row1 * DIM;
#pragma unroll 4
        for (int k0 = 0; k0 < DIM; k0 += 4) {
            const int kk = k0 + half * 2;
            v2f b0, b1;
            b0.x = W2[kk * DIM + n0];
            b0.y = W2[(kk + 1) * DIM + n0];
            b1.x = W2[kk * DIM + n0 + 16];
            b1.y = W2[(kk + 1) * DIM + n0 + 16];
            v2f a0;
            a0.x = brow0[kk];
            a0.y = brow0[kk + 1];
            acc00 = __builtin_amdgcn_wmma_f32_16x16x4_f32(false, a0, false, b0,
                                                          (short)0, acc00, false, false);
            acc01 = __builtin_amdgcn_wmma_f32_16x16x4_f32(false, a0, false, b1,
                                                          (short)0, acc01, false, false);
            if constexpr (HAS2) {
                v2f a1;
                a1.x = brow1[kk];
                a1.y = brow1[kk + 1];
                acc10 = __builtin_amdgcn_wmma_f32_16x16x4_f32(false, a1, false, b0,
                                                              (short)0, acc10, false, false);
                acc11 = __builtin_amdgcn_wmma_f32_16x16x4_f32(false, a1, false, b1,
                                                              (short)0, acc11, false, false);
            }
        }
    }

    // epilogue: C/D layout -> element (M = v + 8*half, N = l16) per tile
    const int colBase = wv * 32;
    const int c0i = colBase + l16;
    const int c1i = colBase + 16 + l16;
    const float bb0 = bias[c0i];
    const float bb1 = bias[c1i];
#pragma unroll
    for (int v = 0; v < 8; ++v) {
        const int r0 = base0 + half * 8 + v;
        float* p0 = C + (size_t)r0 * DIM;
        const float o00 = acc00[v] + bb0;
        const float o01 = acc01[v] + bb1;
        if constexpr (ACCUM) { p0[c0i] += o00; p0[c1i] += o01; }
        else                 { p0[c0i]  = o00; p0[c1i]  = o01; }
    }
    if constexpr (HAS2) {
#pragma unroll
        for (int v = 0; v < 8; ++v) {
            const int r1 = base1 + half * 8 + v;
            float* p1 = C + (size_t)r1 * DIM;
            const float o10 = acc10[v] + bb0;
            const float o11 = acc11[v] + bb1;
            if constexpr (ACCUM) { p1[c0i] += o10; p1[c1i] += o11; }
            else                 { p1[c0i]  = o10; p1[c1i]  = o11; }
        }
    }
}

template <bool FUSED, bool ACCUM>
__global__ __launch_bounds__(128)
void gemm32_wmma(const float* __restrict__ A1, const float* __restrict__ cnt,
                 const float* __restrict__ A2, const float* __restrict__ W1,
                 const float* __restrict__ W2, const float* __restrict__ bias,
                 float* __restrict__ C, int nrows) {
    const int base0 = blockIdx.x * 32;
    if (base0 + 32 <= nrows) {   // block-uniform dispatch, hoisted out of loops
        gemm_tile_body<FUSED, ACCUM, true>(A1, cnt, A2, W1, W2, bias, C, base0);
    } else {
        gemm_tile_body<FUSED, ACCUM, false>(A1, cnt, A2, W1, W2, bias, C, base0);
    }
}

// ---------------- relu + optional residual, in place on out buffer ----------
__global__ void relu_res_kernel(float4* __restrict__ o,
                                const float4* __restrict__ prev,
                                int addRes, size_t n4) {
    size_t i = (size_t)blockIdx.x * blockDim.x + threadIdx.x;
    if (i >= n4) return;
    float4 v = o[i];
    v.x = fmaxf(v.x, 0.f);
    v.y = fmaxf(v.y, 0.f);
    v.z = fmaxf(v.z, 0.f);
    v.w = fmaxf(v.w, 0.f);
    if (addRes) {
        float4 p = prev[i];
        v.x += p.x; v.y += p.y; v.z += p.z; v.w += p.w;
    }
    o[i] = v;
}

// ---------------- column-sum reduction (for mean pooling) ----------------
__global__ void colsum_kernel(const float* __restrict__ x, int n,
                              float* __restrict__ colsum) {
    int col = threadIdx.x;  // 128 threads
    float acc = 0.f;
    for (int r = blockIdx.x; r < n; r += gridDim.x)
        acc += x[(size_t)r * DIM + col];
    atomicAdd(&colsum[col], acc);
}

// ---------------- final pooling head ----------------
// pooled_t = mean(x_t) @ lin2_w + lin2_b ; xout = mean_t(pooled_t @ lin_w + lin_b)
__global__ void final_pool_kernel(const float* __restrict__ colsum,
                                  const float* __restrict__ lin2w,
                                  const float* __restrict__ lin2b,
                                  const float* __restrict__ linw,
                                  const float* __restrict__ linb,
                                  float* __restrict__ xout) {
    __shared__ float pooled[7][DIM];
    int j = threadIdx.x;  // 128 threads
    for (int t = 0; t < 7; ++t) {
        const float invN = (t == 0) ? (1.0f / (float)N_EL) : (1.0f / (float)N_GRPN);
        float s = 0.f;
        for (int k = 0; k < DIM; ++k)
            s += colsum[t * DIM + k] * invN * lin2w[k * DIM + j];
        pooled[t][j] = s + lin2b[j];
    }
    __syncthreads();
    float acc = 0.f;
    for (int t = 0; t < 7; ++t) {
        float s = 0.f;
        for (int k = 0; k < DIM; ++k)
            s += pooled[t][k] * linw[k * DIM + j];
        acc += s + linb[j];
    }
    xout[j] = acc * (1.0f / 7.0f);
}

// ---------------- host driver ----------------

extern "C" void kernel_launch(void* const* d_in, const int* in_sizes, int n_in,
                              void* d_out, int out_size, void* d_ws, size_t ws_size,
                              hipStream_t stream) {
    (void)in_sizes; (void)n_in; (void)out_size; (void)ws_size;

    const float* Wl    = (const float*)d_in[19];
    const float* Wr    = (const float*)d_in[20];
    const float* bvec  = (const float*)d_in[21];
    const float* lin2w = (const float*)d_in[22];
    const float* lin2b = (const float*)d_in[23];
    const float* linw  = (const float*)d_in[24];
    const float* linb  = (const float*)d_in[25];
    float* out = (float*)d_out;

    const size_t X_EL    = (size_t)N_EL * DIM;        // 25,600,000
    const size_t X_GR    = (size_t)N_GRPN * DIM;      // 6,400,000
    const size_t X_TOTAL = X_EL + 6 * X_GR;           // 64,000,000

    float* bufA   = (float*)d_ws;
    float* bufB   = bufA + X_TOTAL;
    float* msum   = bufB + X_TOTAL;
    float* cnt    = msum + X_EL;
    float* colsum = cnt + N_EL;

    size_t toff[7];
    toff[0] = 0;
    for (int t = 1; t < 7; ++t) toff[t] = X_EL + (size_t)(t - 1) * X_GR;
    const int tN[7] = {N_EL, N_GRPN, N_GRPN, N_GRPN, N_GRPN, N_GRPN, N_GRPN};

    // stage inputs into ping buffer
    for (int t = 0; t < 7; ++t)
        hipMemcpyAsync(bufA + toff[t], d_in[t],
                       (size_t)tN[t] * DIM * sizeof(float),
                       hipMemcpyDeviceToDevice, stream);

    const int relSrc[12] = {0, 0, 0, 0, 0, 0, 1, 2, 3, 4, 5, 6};
    const int relDst[12] = {1, 2, 3, 4, 5, 6, 0, 0, 0, 0, 0, 0};

    for (int l = 0; l < 4; ++l) {
        hipMemsetAsync(bufB, 0, X_TOTAL * sizeof(float), stream);
        for (int r = 0; r < 12; ++r) {
            const int st = relSrc[r], dt = relDst[r];
            const int nd = tN[dt];
            const int* edge = (const int*)d_in[7 + r];

            hipMemsetAsync(msum, 0, (size_t)nd * DIM * sizeof(float), stream);
            hipMemsetAsync(cnt, 0, (size_t)nd * sizeof(float), stream);
            scatter_cnt_kernel<<<(NE + 255) / 256, 256, 0, stream>>>(edge, cnt);
            scatter_feat_kernel<<<(NE * 32 + 255) / 256, 256, 0, stream>>>(
                edge, bufA + toff[st], msum);

            const float* w1 = Wl + (size_t)(l * 12 + r) * DIM * DIM;
            const float* w2 = Wr + (size_t)(l * 12 + r) * DIM * DIM;
            const float* bb = bvec + (size_t)(l * 12 + r) * DIM;
            gemm32_wmma<true, true><<<(nd + 31) / 32, 128, 0, stream>>>(
                msum, cnt, bufA + toff[dt], w1, w2, bb, bufB + toff[dt], nd);
        }
        relu_res_kernel<<<(int)((X_TOTAL / 4 + 255) / 256), 256, 0, stream>>>(
            (float4*)bufB, (const float4*)bufA, (l & 1), X_TOTAL / 4);
        float* tmp = bufA; bufA = bufB; bufB = tmp;
    }

    // final head
    hipMemsetAsync(colsum, 0, 7 * DIM * sizeof(float), stream);
    for (int t = 0; t < 7; ++t)
        colsum_kernel<<<512, 128, 0, stream>>>(bufA + toff[t], tN[t],
                                               colsum + t * DIM);
    for (int t = 1; t < 7; ++t)
        gemm32_wmma<false, false><<<(N_GRPN + 31) / 32, 128, 0, stream>>>(
            bufA + toff[t], nullptr, nullptr, lin2w, nullptr, lin2b,
            out + 128 + (size_t)(t - 1) * X_GR, N_GRPN);
    final_pool_kernel<<<1, 128, 0, stream>>>(colsum, lin2w, lin2b, linw, linb, out);
}